// SGPN_62792421867673
// MI455X (gfx1250) — compile-verified
//
#include <hip/hip_runtime.h>
#include <hip/hip_bf16.h>

typedef __attribute__((ext_vector_type(2))) float v2f;
typedef __attribute__((ext_vector_type(8))) float v8f;

#define NPTS   8192
#define NBATCH 2
#define NCH    64
#define NTILE  (NPTS / 16)   // 512 tiles of 16 along N

// ---------------------------------------------------------------------------
// Kernel 1: Fsim = W @ x + b  (3x64 * [B,64,N]),  r = ||Fsim||^2, zero acc.
// Memory-bound (4MB read); one thread per (batch, point).
// ---------------------------------------------------------------------------
__global__ void sgpn_feat(const float* __restrict__ x,   // [B,64,N]
                          const float* __restrict__ W,   // [3,64]
                          const float* __restrict__ bias,// [3]
                          float* __restrict__ F,         // ws: [B,3,N]
                          float* __restrict__ r,         // ws: [B,N]
                          float* __restrict__ acc) {     // ws: [1]
  int idx = blockIdx.x * blockDim.x + threadIdx.x;
  if (idx == 0) acc[0] = 0.0f;
  if (idx >= NBATCH * NPTS) return;
  int b = idx / NPTS;
  int n = idx - b * NPTS;
  const float* xb = x + (size_t)b * NCH * NPTS + n;
  float f0 = bias[0], f1 = bias[1], f2 = bias[2];
#pragma unroll 8
  for (int c = 0; c < NCH; ++c) {
    float xv = xb[(size_t)c * NPTS];
    f0 = fmaf(W[c], xv, f0);
    f1 = fmaf(W[NCH + c], xv, f1);
    f2 = fmaf(W[2 * NCH + c], xv, f2);
  }
  F[((size_t)b * 3 + 0) * NPTS + n] = f0;
  F[((size_t)b * 3 + 1) * NPTS + n] = f1;
  F[((size_t)b * 3 + 2) * NPTS + n] = f2;
  r[(size_t)b * NPTS + n] = f0 * f0 + f1 * f1 + f2 * f2;
}

// ---------------------------------------------------------------------------
// Kernel 2: pairwise Gram via V_WMMA_F32_16X16X4_F32 (K=3 padded to 4),
// branchless fused distance/mask/loss epilogue (v_rcp + 1 Newton step),
// wave32 shuffle reduce, one atomicAdd per wave.
// One wave (32 threads) per (batch, i-tile); loops over all 512 j-tiles.
// ---------------------------------------------------------------------------
__device__ __forceinline__ float tile_term(float dot, float ri, float rj,
                                           int gi, int gj, int ti, int tj) {
  float D = fmaxf(fmaf(-2.0f, dot, ri + rj), 0.0f);
  float x = D + 1.0f;                       // x >= 1: rcp is safe
  float y = __builtin_amdgcn_rcpf(x);       // v_rcp_f32 (~1 ulp)
  y = y * fmaf(-x, y, 2.0f);                // Newton: y ~= 1/x (~0.5 ulp)
  float invsq = y * y;                      // 1/(D+1)^2
  // same-group mask: diag forced 1; label==-1 invalid (never same)
  bool same = (gi == gj) | ((ti == tj) & (ti >= 0));
  // ALPHA=2: term = 2*diff*invsq + same*D  (branchless select)
  return same ? D : 2.0f * invsq;
}

__global__ void sgpn_pair(const float* __restrict__ F,   // [B,3,N]
                          const float* __restrict__ r,   // [B,N]
                          const int*   __restrict__ tgt, // [B,N]
                          float* __restrict__ acc) {
  const int wid = blockIdx.x;            // 0 .. NBATCH*NTILE-1
  const int b   = wid >> 9;              // /512
  const int it  = wid & (NTILE - 1);
  const int i0  = it * 16;
  const int lane = threadIdx.x;          // 0..31 (wave32)
  const int hi   = lane >> 4;            // 0: K={0,1}, 1: K={2,3(=0)}
  const int lm   = lane & 15;

  const float* F0 = F + ((size_t)b * 3 + 0) * NPTS;
  const float* F1 = F + ((size_t)b * 3 + 1) * NPTS;
  const float* F2 = F + ((size_t)b * 3 + 2) * NPTS;
  const float* rb = r + (size_t)b * NPTS;
  const int*   tb = tgt + (size_t)b * NPTS;

  // A-matrix 16x4 f32: lane<16 -> (F0[i],F1[i]) ; lane>=16 -> (F2[i], 0).
  // Load all channels unconditionally (L2-resident), select branchlessly.
  {
    // nothing
  }
  float a0 = F0[i0 + lm], a1 = F1[i0 + lm], a2 = F2[i0 + lm];
  v2f a;
  a.x = hi ? a2   : a0;
  a.y = hi ? 0.0f : a1;

  // Per-lane row metadata for the 8 C/D rows this lane owns: M = v + 8*hi
  float ri[8]; int ti[8]; int gi[8];
#pragma unroll
  for (int v = 0; v < 8; ++v) {
    int m = i0 + v + 8 * hi;
    gi[v] = m;
    ri[v] = rb[m];
    ti[v] = tb[m];
  }

  float sum = 0.0f;
#pragma unroll 2
  for (int jt = 0; jt < NTILE; ++jt) {
    const int j0 = jt * 16;
    // B-matrix 4x16 f32, mirrored layout of A (N in place of M).
    float b0 = F0[j0 + lm], b1 = F1[j0 + lm], b2 = F2[j0 + lm];
    v2f bb;
    bb.x = hi ? b2   : b0;
    bb.y = hi ? 0.0f : b1;
    const float rj = rb[j0 + lm];
    const int   tj = tb[j0 + lm];
    const int   gj = j0 + lm;

    v8f c = {};
    // D = A x B + 0 : dot[i][j] = sum_k F[k][i]*F[k][j]  (exact f32)
    c = __builtin_amdgcn_wmma_f32_16x16x4_f32(
        /*neg_a=*/false, a, /*neg_b=*/false, bb,
        /*c_mod=*/(short)0, c, /*reuse_a=*/false, /*reuse_b=*/false);

#pragma unroll
    for (int v = 0; v < 8; ++v)
      sum += tile_term(c[v], ri[v], rj, gi[v], gj, ti[v], tj);
  }

  // wave32 reduction
#pragma unroll
  for (int s = 16; s > 0; s >>= 1)
    sum += __shfl_down(sum, s, 32);
  if (lane == 0) atomicAdd(acc, sum);
}

// ---------------------------------------------------------------------------
// Kernel 3: final scale -> mean of 100*(...) over B*N*N elements.
// ---------------------------------------------------------------------------
__global__ void sgpn_final(const float* __restrict__ acc, float* __restrict__ out) {
  const float scale = 100.0f / (float)((long long)NBATCH * NPTS * NPTS);
  out[0] = acc[0] * scale;
}

extern "C" void kernel_launch(void* const* d_in, const int* in_sizes, int n_in,
                              void* d_out, int out_size, void* d_ws, size_t ws_size,
                              hipStream_t stream) {
  const float* l0   = (const float*)d_in[0];  // [2,64,8192]
  const float* W    = (const float*)d_in[1];  // [3,64]
  const float* bias = (const float*)d_in[2];  // [3]
  const int*   tgt  = (const int*)d_in[3];    // [2,8192]

  float* Fbuf = (float*)d_ws;                       // 6*8192 floats
  float* rbuf = Fbuf + (size_t)NBATCH * 3 * NPTS;   // 2*8192 floats
  float* acc  = rbuf + (size_t)NBATCH * NPTS;       // 1 float

  int nthreads = NBATCH * NPTS;
  sgpn_feat<<<(nthreads + 255) / 256, 256, 0, stream>>>(l0, W, bias, Fbuf, rbuf, acc);
  sgpn_pair<<<NBATCH * NTILE, 32, 0, stream>>>(Fbuf, rbuf, tgt, acc);
  sgpn_final<<<1, 1, 0, stream>>>(acc, (float*)d_out);
}